// SS2D_13228499272330
// MI455X (gfx1250) — compile-verified
//
#include <hip/hip_runtime.h>
#include <hip/hip_bf16.h>

// ---- SS2D dims ----
#define BB 4
#define HH 64
#define WW 64
#define LL 4096          // H*W
#define DM 96            // d_model
#define DI 192           // d_inner
#define NS 16            // d_state
#define DR 6             // dt_rank
#define KK 4             // scan directions
#define CPAD 48          // padded (dt_rank + 2*d_state) = 38 -> 48

typedef __attribute__((ext_vector_type(2))) float v2f;
typedef __attribute__((ext_vector_type(8))) float v8f;

__device__ __forceinline__ v8f wmma_f32_4(v2f a, v2f b, v8f c) {
    // V_WMMA_F32_16X16X4_F32 : D = A(16x4) * B(4x16) + C(16x16), fp32
    return __builtin_amdgcn_wmma_f32_16x16x4_f32(
        /*neg_a=*/false, a, /*neg_b=*/false, b,
        /*c_mod=*/(short)0, c, /*reuse_a=*/false, /*reuse_b=*/false);
}

__device__ __forceinline__ v2f ldv2(const float* p) {
    return *(const v2f*)p;   // 8B-aligned by construction
}

// cross-scan index map: position p in direction k -> row-major (h*W+w) index
__device__ __forceinline__ int perm_idx(int k, int p) {
    int q = (k & 2) ? (LL - 1 - p) : p;
    if (k & 1) q = (q & 63) * 64 + (q >> 6);   // transpose map (H==W==64)
    return q;
}

// ---------------- 1) in_proj: X (16384x96) @ W^T (96x384) -> XZ ----------------
__global__ void k_gemm_inproj(const float* __restrict__ X,
                              const float* __restrict__ W,
                              float* __restrict__ XZ) {
    const int NT = (2 * DI) / 16;            // 24 N-tiles
    const int MT = (BB * LL) / 16;           // 1024 M-tiles
    int wave = blockIdx.x * (blockDim.x >> 5) + (threadIdx.x >> 5);
    int lane = threadIdx.x & 31;
    if (wave >= MT * NT) return;
    int mt = wave / NT, nt = wave % NT;
    int m  = mt * 16 + (lane & 15);
    int n  = nt * 16 + (lane & 15);
    int kh = (lane >> 4) * 2;                // lanes 16-31 carry K=2,3
    const float* Ap = X + (size_t)m * DM + kh;   // 8B aligned (m*96+kh even)
    const float* Bp = W + (size_t)n * DM + kh;   // B[k,n] = W[n,k]
    v8f acc = {};
#pragma unroll
    for (int k0 = 0; k0 < DM; k0 += 4)
        acc = wmma_f32_4(ldv2(Ap + k0), ldv2(Bp + k0), acc);
    int row0 = (lane >> 4) * 8;
    int col  = nt * 16 + (lane & 15);
#pragma unroll
    for (int r = 0; r < 8; ++r)
        XZ[(size_t)(mt * 16 + row0 + r) * (2 * DI) + col] = acc[r];
}

// ---------------- 2) depthwise 3x3 conv + SiLU -> XC (b, l, d) ----------------
__global__ void k_conv(const float* __restrict__ XZ, const float* __restrict__ CW,
                       const float* __restrict__ CB, float* __restrict__ XC) {
    int idx = blockIdx.x * blockDim.x + threadIdx.x;
    if (idx >= BB * LL * DI) return;
    int d = idx % DI;
    int l = (idx / DI) % LL;
    int b = idx / (DI * LL);
    int h = l >> 6, w = l & 63;
    float acc = CB[d];
#pragma unroll
    for (int dy = -1; dy <= 1; ++dy)
#pragma unroll
        for (int dx = -1; dx <= 1; ++dx) {
            int hh = h + dy, ww = w + dx;
            if (hh >= 0 && hh < HH && ww >= 0 && ww < WW)
                acc += XZ[(size_t)(b * LL + hh * WW + ww) * (2 * DI) + d] *
                       CW[d * 9 + (dy + 1) * 3 + (dx + 1)];
        }
    XC[(size_t)(b * LL + l) * DI + d] = acc / (1.f + __expf(-acc));  // SiLU
}

// ------- 3) x_proj per (b,k): A = xs^T (4096x192) @ Wk^T (192x38->48) --------
__global__ void k_gemm_xproj(const float* __restrict__ XC,
                             const float* __restrict__ XPW,
                             float* __restrict__ XDBL) {
    const int NT = CPAD / 16;                // 3
    const int MT = LL / 16;                  // 256
    int wave = blockIdx.x * (blockDim.x >> 5) + (threadIdx.x >> 5);
    int lane = threadIdx.x & 31;
    if (wave >= BB * KK * MT * NT) return;
    int nt = wave % NT; int t = wave / NT;
    int mt = t % MT;    t /= MT;
    int k  = t % KK;    int b = t / KK;
    int mrow = mt * 16 + (lane & 15);
    int lsrc = perm_idx(k, mrow);
    int n    = nt * 16 + (lane & 15);
    int kh   = (lane >> 4) * 2;
    bool nv  = (n < 38);
    const float* Ap = XC + (size_t)(b * LL + lsrc) * DI + kh;
    // guard OOB rows of padded B with a safe pointer (result masked to 0)
    const float* Bp = XPW + (size_t)k * 38 * DI + (size_t)(nv ? n : 0) * DI + kh;
    v8f acc = {};
#pragma unroll
    for (int k0 = 0; k0 < DI; k0 += 4) {
        v2f a  = ldv2(Ap + k0);
        v2f bb = ldv2(Bp + k0);
        if (!nv) { bb[0] = 0.f; bb[1] = 0.f; }
        acc = wmma_f32_4(a, bb, acc);
    }
    int row0 = (lane >> 4) * 8;
    int col  = nt * 16 + (lane & 15);
    float* outp = XDBL + (size_t)((b * KK + k) * LL) * CPAD;
#pragma unroll
    for (int r = 0; r < 8; ++r)
        outp[(size_t)(mt * 16 + row0 + r) * CPAD + col] = acc[r];
}

// ----- 4) selective scan: 1 block per (b,k) chain, 1 thread per channel d -----
__global__ void __launch_bounds__(DI)
k_scan(const float* __restrict__ XC, const float* __restrict__ XDBL,
       const float* __restrict__ DTW, const float* __restrict__ DTB,
       const float* __restrict__ ALOG, const float* __restrict__ DSK,
       float* __restrict__ YS) {
    int bk = blockIdx.x;                 // 0..15
    int b  = bk >> 2, k = bk & 3;
    int d  = threadIdx.x;                // 0..191
    __shared__ float sh[2][CPAD];        // double buffer -> 1 barrier / step

    float A[NS];
#pragma unroll
    for (int n = 0; n < NS; ++n)
        A[n] = -__expf(ALOG[(size_t)(k * DI + d) * NS + n]);
    float hst[NS];
#pragma unroll
    for (int n = 0; n < NS; ++n) hst[n] = 0.f;
    float dtw[DR];
#pragma unroll
    for (int r = 0; r < DR; ++r) dtw[r] = DTW[(size_t)(k * DI + d) * DR + r];
    const float bias = DTB[k * DI + d];
    const float Dsk  = DSK[k * DI + d];

    const float* xdbl = XDBL + (size_t)bk * LL * CPAD;
    float* ys = YS + (size_t)bk * LL * DI;

    if (threadIdx.x < CPAD) sh[0][threadIdx.x] = xdbl[threadIdx.x];
    __syncthreads();

    for (int l = 0; l < LL; ++l) {
        int cur = l & 1;
        // prefetch next step's projections into the other buffer
        if (threadIdx.x < CPAD && (l + 1) < LL)
            sh[cur ^ 1][threadIdx.x] = xdbl[(size_t)(l + 1) * CPAD + threadIdx.x];

        float u = XC[(size_t)(b * LL + perm_idx(k, l)) * DI + d];
        float dt = bias;
#pragma unroll
        for (int r = 0; r < DR; ++r) dt += dtw[r] * sh[cur][r];
        float delta = (dt > 20.f) ? dt : __logf(1.f + __expf(dt));   // softplus
        float du = delta * u;
        float y  = Dsk * u;
#pragma unroll
        for (int n = 0; n < NS; ++n) {
            hst[n] = hst[n] * __expf(delta * A[n]) + du * sh[cur][DR + n];
            y += hst[n] * sh[cur][DR + NS + n];
        }
        ys[(size_t)l * DI + d] = y;
        __syncthreads();   // next-buffer writes visible; cur-buffer reads done
    }
}

// -------- 5) cross-merge + LayerNorm + SiLU gate, one block per token --------
__global__ void __launch_bounds__(DI)
k_merge_ln_gate(const float* __restrict__ YS, const float* __restrict__ XZ,
                const float* __restrict__ LNW, const float* __restrict__ LNB,
                float* __restrict__ YG) {
    int t = blockIdx.x;                  // b*4096 + l
    int b = t >> 12, l = t & (LL - 1);
    int d = threadIdx.x;
    int lt = (l & 63) * 64 + (l >> 6);   // transpose-mapped position
    size_t base = (size_t)b * KK * LL * DI;
    float y = YS[base + ((size_t)(0 * LL + l)            * DI) + d]
            + YS[base + ((size_t)(1 * LL + lt)           * DI) + d]
            + YS[base + ((size_t)(2 * LL + (LL - 1 - l)) * DI) + d]
            + YS[base + ((size_t)(3 * LL + (LL - 1 - lt))* DI) + d];

    __shared__ float r1[DI], r2[DI];
    __shared__ float s_mu, s_inv;
    r1[d] = y; r2[d] = y * y;
    __syncthreads();
#pragma unroll
    for (int s = DI / 2; s >= 3; s >>= 1) {   // 96,48,24,12,6,3
        if (d < s) { r1[d] += r1[d + s]; r2[d] += r2[d + s]; }
        __syncthreads();
    }
    if (d == 0) {
        float sum  = r1[0] + r1[1] + r1[2];
        float sum2 = r2[0] + r2[1] + r2[2];
        float mu   = sum / (float)DI;
        float var  = sum2 / (float)DI - mu * mu;
        s_mu  = mu;
        s_inv = rsqrtf(var + 1e-5f);
    }
    __syncthreads();
    float yn = (y - s_mu) * s_inv * LNW[d] + LNB[d];
    float z  = XZ[(size_t)t * (2 * DI) + DI + d];
    float g  = z / (1.f + __expf(-z));        // SiLU
    YG[(size_t)t * DI + d] = yn * g;
}

// ---------------- 6) out_proj: YG (16384x192) @ W^T (192x96) -----------------
__global__ void k_gemm_outproj(const float* __restrict__ YG,
                               const float* __restrict__ W,
                               float* __restrict__ OUT) {
    const int NT = DM / 16;                  // 6
    const int MT = (BB * LL) / 16;           // 1024
    int wave = blockIdx.x * (blockDim.x >> 5) + (threadIdx.x >> 5);
    int lane = threadIdx.x & 31;
    if (wave >= MT * NT) return;
    int mt = wave / NT, nt = wave % NT;
    int m  = mt * 16 + (lane & 15);
    int n  = nt * 16 + (lane & 15);
    int kh = (lane >> 4) * 2;
    const float* Ap = YG + (size_t)m * DI + kh;
    const float* Bp = W  + (size_t)n * DI + kh;
    v8f acc = {};
#pragma unroll
    for (int k0 = 0; k0 < DI; k0 += 4)
        acc = wmma_f32_4(ldv2(Ap + k0), ldv2(Bp + k0), acc);
    int row0 = (lane >> 4) * 8;
    int col  = nt * 16 + (lane & 15);
#pragma unroll
    for (int r = 0; r < 8; ++r)
        OUT[(size_t)(mt * 16 + row0 + r) * DM + col] = acc[r];
}

extern "C" void kernel_launch(void* const* d_in, const int* in_sizes, int n_in,
                              void* d_out, int out_size, void* d_ws, size_t ws_size,
                              hipStream_t stream) {
    (void)in_sizes; (void)n_in; (void)out_size; (void)ws_size;
    const float* x      = (const float*)d_in[0];   // (4,64,64,96)
    const float* ipw    = (const float*)d_in[1];   // (384,96)
    const float* convw  = (const float*)d_in[2];   // (192,1,3,3)
    const float* convb  = (const float*)d_in[3];   // (192,)
    const float* xpw    = (const float*)d_in[4];   // (4,38,192)
    const float* dtw    = (const float*)d_in[5];   // (4,192,6)
    const float* dtb    = (const float*)d_in[6];   // (4,192)
    const float* alog   = (const float*)d_in[7];   // (768,16)
    const float* dsk    = (const float*)d_in[8];   // (768,)
    const float* lnw    = (const float*)d_in[9];   // (192,)
    const float* lnb    = (const float*)d_in[10];  // (192,)
    const float* opw    = (const float*)d_in[11];  // (96,192)
    float* out = (float*)d_out;

    // workspace carve (fp32): ~113 MB total
    float* xz   = (float*)d_ws;                        // 16384*384
    float* xc   = xz   + (size_t)BB * LL * 2 * DI;     // 16384*192
    float* xdbl = xc   + (size_t)BB * LL * DI;         // 16*4096*48
    float* ys   = xdbl + (size_t)BB * KK * LL * CPAD;  // 16*4096*192
    float* yg   = ys   + (size_t)BB * KK * LL * DI;    // 16384*192

    {   // in_proj: 1024*24 wave-tiles, 8 waves/block
        int waves = ((BB * LL) / 16) * ((2 * DI) / 16);
        k_gemm_inproj<<<(waves + 7) / 8, 256, 0, stream>>>(x, ipw, xz);
    }
    {   // depthwise conv + silu
        int nel = BB * LL * DI;
        k_conv<<<(nel + 255) / 256, 256, 0, stream>>>(xz, convw, convb, xc);
    }
    {   // x_proj (per b,k): 4*4*256*3 wave-tiles
        int waves = BB * KK * (LL / 16) * (CPAD / 16);
        k_gemm_xproj<<<(waves + 7) / 8, 256, 0, stream>>>(xc, xpw, xdbl);
    }
    {   // selective scan: 16 sequential chains
        k_scan<<<BB * KK, DI, 0, stream>>>(xc, xdbl, dtw, dtb, alog, dsk, ys);
    }
    {   // merge + LN + gate
        k_merge_ln_gate<<<BB * LL, DI, 0, stream>>>(ys, xz, lnw, lnb, yg);
    }
    {   // out_proj
        k_gemm_outproj<<<(((BB * LL) / 16) * (DM / 16) + 7) / 8, 256, 0, stream>>>(yg, opw, out);
    }
}